// ScaledDotProductAttention_386547056800
// MI455X (gfx1250) — compile-verified
//
#include <hip/hip_runtime.h>
#include <math.h>

// CDNA5 / gfx1250: wave32, V_WMMA_F32_16X16X4_F32 (fp32 matrix pipe).
typedef __attribute__((ext_vector_type(2))) float v2f;
typedef __attribute__((ext_vector_type(8))) float v8f;

#define BATCH   32
#define SEQ     2048
#define DHEAD   64
#define QTILE   16
#define SROW    2052          // padded LDS row stride (floats): +4 avoids bank conflicts
#define NKT     (SEQ / 16)    // 128 k-tiles
#define NWAVES  4
#define TPB     (NWAVES * 32) // 128 threads = 4 wave32

// Deterministic stateless dropout hash (splitmix64-style). The JAX threefry
// stream is not bit-reproducible here; determinism across replays is what the
// harness requires, which this provides.
__device__ __forceinline__ float dropout_u01(unsigned long long gi) {
    unsigned long long x = gi + 42ull;
    x ^= x >> 33; x *= 0xff51afd7ed558ccdULL;
    x ^= x >> 33; x *= 0xc4ceb9fe1a85ec53ULL;
    x ^= x >> 33;
    return (float)(unsigned int)x * 2.3283064365386963e-10f; // [0,1)
}

__global__ __launch_bounds__(TPB)
void sdpa_wmma_f32_kernel(const float* __restrict__ Q,
                          const float* __restrict__ K,
                          const float* __restrict__ V,
                          const int*   __restrict__ mask,
                          float*       __restrict__ out) {
    extern __shared__ float smem[];
    float* S_lds  = smem;                       // QTILE * SROW floats (scores -> probs)
    float* red    = smem + QTILE * SROW;        // 16 * 8 partials
    float* rowmax = red + QTILE * 8;            // 16
    float* rowinv = rowmax + QTILE;             // 16

    const int bid  = blockIdx.x;
    const int b    = bid >> 7;                  // / (SEQ/QTILE)
    const int q0   = (bid & 127) * QTILE;

    const int tid  = threadIdx.x;
    const int wave = tid >> 5;
    const int lane = tid & 31;
    const int half = lane >> 4;                 // 0: lanes 0-15, 1: lanes 16-31
    const int l16  = lane & 15;

    // ---- Q A-fragments in registers: 16 chunks of K=4 (v2f each, WMMA layout) ----
    // A 16x4 f32 layout: lanes 0-15 hold M=lane,K={0,1}; lanes 16-31 hold M=lane-16,K={2,3}.
    v2f qa[16];
    {
        const float* qp = Q + (size_t)(b * SEQ + q0 + l16) * DHEAD + 2 * half;
        #pragma unroll
        for (int c = 0; c < 16; ++c) {
            qa[c].x = qp[4 * c + 0];
            qa[c].y = qp[4 * c + 1];
        }
    }

    // ---- Phase 1: scores = (Q K^T) * 1/sqrt(64), masked, into LDS ----
    for (int kt = wave; kt < NKT; kt += NWAVES) {          // wave-uniform loop
        const int k0 = kt * 16;
        // B 4x16 layout: lanes 0-15 hold K={0,1},N=lane; lanes 16-31 hold K={2,3},N=lane-16.
        const float* kp = K + (size_t)(b * SEQ + k0 + l16) * DHEAD + 2 * half;
        v8f acc0 = {}, acc1 = {};                // two independent chains for ILP
        #pragma unroll
        for (int c = 0; c < 16; c += 2) {
            v2f kb0; kb0.x = kp[4 * c + 0]; kb0.y = kp[4 * c + 1];
            v2f kb1; kb1.x = kp[4 * c + 4]; kb1.y = kp[4 * c + 5];
            acc0 = __builtin_amdgcn_wmma_f32_16x16x4_f32(
                       false, qa[c],     false, kb0, (short)0, acc0, false, false);
            acc1 = __builtin_amdgcn_wmma_f32_16x16x4_f32(
                       false, qa[c + 1], false, kb1, (short)0, acc1, false, false);
        }
        const v8f acc = acc0 + acc1;
        // C/D layout: VGPR j -> row m = j + 8*half, col n = l16.
        const int col = k0 + l16;
        #pragma unroll
        for (int j = 0; j < 8; ++j) {
            const int m = j + 8 * half;
            float s = acc[j] * 0.125f;
            // Mask is streamed exactly once (512 MB total): non-temporal so it
            // does not evict the reusable Q/K/V working set from the 192 MB L2.
            const int mk = __builtin_nontemporal_load(
                &mask[(size_t)(b * SEQ + q0 + m) * SEQ + col]);
            if (mk == 0) s = -INFINITY;
            S_lds[m * SROW + col] = s;
        }
    }
    __syncthreads();

    // ---- Phase 2: row softmax (8 threads per row) ----
    {
        const int r = tid >> 3, j8 = tid & 7;
        float m = -INFINITY;
        for (int c = j8; c < SEQ; c += 8) m = fmaxf(m, S_lds[r * SROW + c]);
        red[r * 8 + j8] = m;
    }
    __syncthreads();
    if (tid < QTILE) {
        float m = red[tid * 8];
        #pragma unroll
        for (int j = 1; j < 8; ++j) m = fmaxf(m, red[tid * 8 + j]);
        rowmax[tid] = m;
    }
    __syncthreads();
    {
        const int r = tid >> 3, j8 = tid & 7;
        const float rm = rowmax[r];
        float s = 0.0f;
        for (int c = j8; c < SEQ; c += 8) {
            const float e = __expf(S_lds[r * SROW + c] - rm);
            S_lds[r * SROW + c] = e;
            s += e;
        }
        red[r * 8 + j8] = s;
    }
    __syncthreads();
    if (tid < QTILE) {
        float s = 0.0f;
        #pragma unroll
        for (int j = 0; j < 8; ++j) s += red[tid * 8 + j];
        rowinv[tid] = 1.0f / s;
    }
    __syncthreads();

    // ---- Phase 3: normalize + dropout; stream attention tile out (NT stores),
    //      keep post-dropout probabilities in LDS for the AV matmul ----
    float* attn = out + (size_t)BATCH * SEQ * DHEAD + (size_t)(b * SEQ + q0) * SEQ;
    for (int idx = tid; idx < QTILE * SEQ; idx += TPB) {
        const int r = idx >> 11;
        const int c = idx & (SEQ - 1);
        const float p = S_lds[r * SROW + c] * rowinv[r];
        const unsigned long long gi =
            (unsigned long long)(b * SEQ + q0 + r) * (unsigned long long)SEQ + c;
        const float pd = (dropout_u01(gi) < 0.8f) ? p * 1.25f : 0.0f;
        S_lds[r * SROW + c] = pd;
        // 512 MB streamed write, never re-read: non-temporal to protect L2.
        __builtin_nontemporal_store(pd, &attn[(size_t)r * SEQ + c]);
    }
    __syncthreads();

    // ---- Phase 4: O = A_d @ V ; each wave owns a 16-wide d-slice ----
    {
        const int d0 = wave * 16;
        v8f acc0 = {}, acc1 = {};                // two independent chains for ILP
        const float* vp   = V + ((size_t)b * SEQ + 2 * half) * DHEAD + d0 + l16;
        const float* srow = S_lds + l16 * SROW + 2 * half;   // A row = l16 for all lanes
        #pragma unroll 4
        for (int c = 0; c < SEQ / 4; c += 2) {
            v2f a0;  a0.x  = srow[4 * c + 0]; a0.y = srow[4 * c + 1];
            v2f a1;  a1.x  = srow[4 * c + 4]; a1.y = srow[4 * c + 5];
            v2f vb0; vb0.x = vp[(size_t)(4 * c + 0) * DHEAD];
                     vb0.y = vp[(size_t)(4 * c + 1) * DHEAD];
            v2f vb1; vb1.x = vp[(size_t)(4 * c + 4) * DHEAD];
                     vb1.y = vp[(size_t)(4 * c + 5) * DHEAD];
            acc0 = __builtin_amdgcn_wmma_f32_16x16x4_f32(
                       false, a0, false, vb0, (short)0, acc0, false, false);
            acc1 = __builtin_amdgcn_wmma_f32_16x16x4_f32(
                       false, a1, false, vb1, (short)0, acc1, false, false);
        }
        const v8f acc = acc0 + acc1;
        #pragma unroll
        for (int j = 0; j < 8; ++j) {
            const int m = j + 8 * half;
            out[(size_t)(b * SEQ + q0 + m) * DHEAD + d0 + l16] = acc[j];
        }
    }
}

extern "C" void kernel_launch(void* const* d_in, const int* in_sizes, int n_in,
                              void* d_out, int out_size, void* d_ws, size_t ws_size,
                              hipStream_t stream) {
    const float* Q    = (const float*)d_in[0];
    const float* K    = (const float*)d_in[1];
    const float* V    = (const float*)d_in[2];
    const int*   mask = (const int*)d_in[3];
    float* out = (float*)d_out;

    const size_t smem_bytes =
        (size_t)(QTILE * SROW + QTILE * 8 + QTILE + QTILE) * sizeof(float);

    // Opt in to large dynamic LDS (idempotent; not a stream op, capture-safe).
    (void)hipFuncSetAttribute((const void*)sdpa_wmma_f32_kernel,
                              hipFuncAttributeMaxDynamicSharedMemorySize,
                              (int)smem_bytes);

    const int grid = BATCH * (SEQ / QTILE);   // 4096 blocks
    sdpa_wmma_f32_kernel<<<grid, TPB, smem_bytes, stream>>>(Q, K, V, mask, out);
}